// VectorQuantizerEMA_2027224564663
// MI455X (gfx1250) — compile-verified
//
#include <hip/hip_runtime.h>
#include <math.h>

typedef __attribute__((ext_vector_type(2))) float v2f;
typedef __attribute__((ext_vector_type(8))) float v8f;

namespace {
constexpr int DIM = 256;
constexpr int KCODES = 1024;
constexpr int NROWS = 16 * 4096;     // 65536 rows
constexpr int ROWS_PER_BLOCK = 128;  // 8 waves x 16 rows

// flat output layout (reference tuple order)
constexpr size_t OFS_LOSS = 0;
constexpr size_t OFS_Q    = 1;
constexpr size_t OFS_PERP = OFS_Q + (size_t)NROWS * DIM;      // 16777217
constexpr size_t OFS_ENC  = OFS_PERP + 1;                      // 16777218
constexpr size_t OFS_DIST = OFS_ENC + (size_t)NROWS * KCODES;  // 83886082
constexpr size_t OFS_IDX  = OFS_DIST + (size_t)NROWS * KCODES; // 150994946
}

// ---------------------------------------------------------------- init
__global__ void vq_init_kernel(unsigned* counts, float* loss_acc) {
  const int t = threadIdx.x;
  counts[t] = 0u;
  if (t == 0) *loss_acc = 0.f;
}

// ---------------------------------------------------------------- ||e_k||^2
__global__ void vq_enorm_kernel(const float* __restrict__ emb,
                                float* __restrict__ enorm) {
  __shared__ float red[256];
  const int r = blockIdx.x, t = threadIdx.x;
  const float v = emb[(size_t)r * DIM + t];
  red[t] = v * v;
  __syncthreads();
  for (int s = 128; s > 0; s >>= 1) {
    if (t < s) red[t] += red[t + s];
    __syncthreads();
  }
  if (t == 0) enorm[r] = red[0];
}

// ---------------------------------------------------------------- distances + argmin (WMMA f32)
__global__ __launch_bounds__(256) void vq_dist_kernel(
    const float* __restrict__ X, const float* __restrict__ E,
    const float* __restrict__ enorm,
    float* __restrict__ dist_out, float* __restrict__ idx_out) {
  __shared__ float Elds[16][260];        // 16 codes x 256, padded stride (260%64==4 -> no bank conflicts)
  __shared__ float en_lds[16];
  __shared__ float xn_lds[ROWS_PER_BLOCK];

  const int t    = threadIdx.x;
  const int wave = t >> 5;
  const int lane = t & 31;
  const int r16  = lane & 15;            // row (A) / col (B,C,D) within tile
  const int h    = lane >> 4;            // half-wave: K-pair select / M+8 select

  const size_t rowbase = (size_t)blockIdx.x * ROWS_PER_BLOCK + (size_t)wave * 16;
  const size_t myrow   = rowbase + r16;

  // A-matrix half-row in registers, laid out exactly as WMMA A fragments need:
  // lane (r16, h) holds X[row][4j + 2h] and X[row][4j + 2h + 1] for j=0..63.
  float xa[128];
  {
    const float* xp = X + myrow * DIM + 2 * h;
    float ss = 0.f;
#pragma unroll
    for (int j = 0; j < 64; ++j) {
      const float2 v = *reinterpret_cast<const float2*>(xp + 4 * j);
      xa[2 * j]     = v.x;
      xa[2 * j + 1] = v.y;
      ss += v.x * v.x + v.y * v.y;
    }
    ss += __shfl_xor(ss, 16);            // combine the two half-rows -> full ||x||^2
    if (h == 0) xn_lds[wave * 16 + r16] = ss;
  }

  float minv[8];
  int   mini[8];
#pragma unroll
  for (int v = 0; v < 8; ++v) { minv[v] = 3.402823e38f; mini[v] = 0; }

  for (int ct = 0; ct < 64; ++ct) {
    const int colbase = ct * 16;
    __syncthreads();                     // also publishes xn_lds on first pass
    {
      // cooperatively stage 16x256 embedding tile into LDS (coalesced float4)
      const int tr = t >> 4;
      const int tc = (t & 15) * 16;
      const float4* src = reinterpret_cast<const float4*>(E + (size_t)(colbase + tr) * DIM + tc);
      float4* dst = reinterpret_cast<float4*>(&Elds[tr][tc]);
#pragma unroll
      for (int q = 0; q < 4; ++q) dst[q] = src[q];
      if (t < 16) en_lds[t] = enorm[colbase + t];
      if (ct + 1 < 64 && t < 16)
        __builtin_prefetch(E + (size_t)(colbase + 16 + t) * DIM, 0, 0);
    }
    __syncthreads();

    v8f acc = {0.f, 0.f, 0.f, 0.f, 0.f, 0.f, 0.f, 0.f};
#pragma unroll
    for (int j = 0; j < 64; ++j) {
      v2f a = {xa[2 * j], xa[2 * j + 1]};
      const float2 bb = *reinterpret_cast<const float2*>(&Elds[r16][4 * j + 2 * h]);
      v2f b = {bb.x, bb.y};
      // D = A(16x4) * B(4x16) + C  ; f32 throughout
      acc = __builtin_amdgcn_wmma_f32_16x16x4_f32(
          false, a, false, b, (short)0, acc, false, false);
    }

    const float en = en_lds[r16];
#pragma unroll
    for (int v = 0; v < 8; ++v) {
      const int lrow = v + 8 * h;        // C/D layout: VGPR v holds rows v / v+8
      const float dval = xn_lds[wave * 16 + lrow] + en - 2.f * acc[v];
      dist_out[(rowbase + lrow) * (size_t)KCODES + colbase + r16] = dval;
      if (dval < minv[v]) { minv[v] = dval; mini[v] = colbase + r16; }
    }
  }

  // argmin across the 16 lanes of each half-wave (xor masks <=8 stay in-half)
#pragma unroll
  for (int v = 0; v < 8; ++v) {
    float bv = minv[v];
    int   bi = mini[v];
#pragma unroll
    for (int m = 8; m >= 1; m >>= 1) {
      const float ov = __shfl_xor(bv, m);
      const int   oi = __shfl_xor(bi, m);
      if (ov < bv || (ov == bv && oi < bi)) { bv = ov; bi = oi; }
    }
    if (r16 == 0) idx_out[rowbase + v + 8 * h] = (float)bi;
  }
}

// ---------------------------------------------------------------- gather + one-hot + loss/hist
__global__ __launch_bounds__(64) void vq_quant_kernel(
    const float* __restrict__ X, const float* __restrict__ E,
    const float* __restrict__ idx_in,
    float* __restrict__ q_out, float* __restrict__ enc_out,
    float* __restrict__ loss_acc, unsigned* __restrict__ counts) {
  const int i = blockIdx.x;
  const int t = threadIdx.x;
  const int idx = (int)idx_in[i];

  const float4 q = reinterpret_cast<const float4*>(E + (size_t)idx * DIM)[t];
  const float4 x = reinterpret_cast<const float4*>(X + (size_t)i * DIM)[t];
  reinterpret_cast<float4*>(q_out + (size_t)i * DIM)[t] = q;  // quantized_st == quantized

  const float dx = q.x - x.x, dy = q.y - x.y, dz = q.z - x.z, dw = q.w - x.w;
  float err = dx * dx + dy * dy + dz * dz + dw * dw;
#pragma unroll
  for (int m = 16; m >= 1; m >>= 1) err += __shfl_xor(err, m);
  __shared__ float wred[2];
  if ((t & 31) == 0) wred[t >> 5] = err;
  __syncthreads();
  if (t == 0) atomicAdd(loss_acc, wred[0] + wred[1]);

  // one-hot encodings row: 64 threads x 16 cols, branch-free float4 stores
  float4* er = reinterpret_cast<float4*>(enc_out + (size_t)i * KCODES) + t * 4;
#pragma unroll
  for (int q4 = 0; q4 < 4; ++q4) {
    const int base = t * 16 + q4 * 4;
    float4 v;
    v.x = (idx == base)     ? 1.f : 0.f;
    v.y = (idx == base + 1) ? 1.f : 0.f;
    v.z = (idx == base + 2) ? 1.f : 0.f;
    v.w = (idx == base + 3) ? 1.f : 0.f;
    er[q4] = v;
  }
  if (t == 0) atomicAdd(&counts[idx], 1u);
}

// ---------------------------------------------------------------- scalars
__global__ __launch_bounds__(1024) void vq_final_kernel(
    const unsigned* __restrict__ counts, const float* __restrict__ loss_acc,
    float* __restrict__ out) {
  const int t = threadIdx.x;
  const float p = (float)counts[t] * (1.f / (float)NROWS);
  float term = p * logf(p + 1e-10f);
#pragma unroll
  for (int m = 16; m >= 1; m >>= 1) term += __shfl_xor(term, m);
  __shared__ float red[32];
  if ((t & 31) == 0) red[t >> 5] = term;
  __syncthreads();
  if (t < 32) {
    float s = red[t];
#pragma unroll
    for (int m = 16; m >= 1; m >>= 1) s += __shfl_xor(s, m);
    if (t == 0) {
      out[OFS_PERP] = expf(-s);
      out[OFS_LOSS] = 0.25f * (*loss_acc) * (1.f / (float)((size_t)NROWS * DIM));
    }
  }
}

// ---------------------------------------------------------------- launch
extern "C" void kernel_launch(void* const* d_in, const int* in_sizes, int n_in,
                              void* d_out, int out_size, void* d_ws, size_t ws_size,
                              hipStream_t stream) {
  const float* X = (const float*)d_in[0];   // inputs  [16,4096,256]
  const float* E = (const float*)d_in[1];   // embedding [1024,256]
  float* out = (float*)d_out;

  float*    enorm    = (float*)d_ws;                          // 4 KB
  unsigned* counts   = (unsigned*)((char*)d_ws + 4096);       // 4 KB
  float*    loss_acc = (float*)((char*)d_ws + 8192);          // 4 B

  float* q_out    = out + OFS_Q;
  float* enc_out  = out + OFS_ENC;
  float* dist_out = out + OFS_DIST;
  float* idx_out  = out + OFS_IDX;

  vq_init_kernel<<<1, 1024, 0, stream>>>(counts, loss_acc);
  vq_enorm_kernel<<<KCODES, 256, 0, stream>>>(E, enorm);
  vq_dist_kernel<<<NROWS / ROWS_PER_BLOCK, 256, 0, stream>>>(X, E, enorm, dist_out, idx_out);
  vq_quant_kernel<<<NROWS, 64, 0, stream>>>(X, E, idx_out, q_out, enc_out, loss_acc, counts);
  vq_final_kernel<<<1, 1024, 0, stream>>>(counts, loss_acc, out);
}